// Discriminator_38199439131010
// MI455X (gfx1250) — compile-verified
//
#include <hip/hip_runtime.h>

// CDNA5 / gfx1250: wave32, WMMA 16x16x4 F32.
// Each wave computes 16 edge dot-products as the diagonal of a 16x16
// A*B^T block via 32 chained V_WMMA_F32_16X16X4_F32 ops (full fp32,
// matching reference numerics). Gather traffic is the minimal
// 2 x 512B of Z per edge; the matrix pipe does the K-reduction so no
// cross-lane shuffle tree is needed.

typedef __attribute__((ext_vector_type(2))) float v2f;
typedef __attribute__((ext_vector_type(8))) float v8f;

#define D_DIM 128
#define EDGES_PER_WAVE 16
#define WAVES_PER_BLOCK 8
#define EDGES_PER_BLOCK (EDGES_PER_WAVE * WAVES_PER_BLOCK)

__global__ __launch_bounds__(256)
void edge_sigmoid_wmma(const float* __restrict__ Z,
                       const int* __restrict__ edges,   // [2, E] int32
                       const float* __restrict__ w,
                       const float* __restrict__ b,
                       float* __restrict__ out,
                       int E) {
    const int lane = threadIdx.x & 31;
    const int wave = threadIdx.x >> 5;
    const int base = (blockIdx.x * WAVES_PER_BLOCK + wave) * EDGES_PER_WAVE;

    // Operand feeding per ISA layout:
    //   A (16x4 f32): lanes 0-15 hold M=lane, K=k..k+1 ; lanes 16-31 hold K=k+2..k+3
    //   B (4x16 f32): lanes 0-15 hold N=lane, K=k..k+1 ; lanes 16-31 hold K=k+2..k+3
    const int eslot = lane & 15;   // which edge of the group this lane feeds
    const int khalf = lane >> 4;   // 0 -> K lo pair, 1 -> K hi pair

    // Clamp (never branch) so EXEC stays all-1s through the WMMA region.
    int eidx = base + eslot;
    if (eidx > E - 1) eidx = E - 1;

    const int srcRow = edges[eidx];       // row 0 of edge_index
    const int dstRow = edges[E + eidx];   // row 1 of edge_index

    const float* __restrict__ aPtr = Z + (size_t)srcRow * D_DIM + khalf * 2;
    const float* __restrict__ bPtr = Z + (size_t)dstRow * D_DIM + khalf * 2;

    // Two accumulators to break the WMMA->WMMA RAW dependency chain.
    v8f acc0 = {};
    v8f acc1 = {};
#pragma unroll
    for (int k = 0; k < D_DIM; k += 8) {
        v2f a0 = *(const v2f*)(aPtr + k);
        v2f b0 = *(const v2f*)(bPtr + k);
        v2f a1 = *(const v2f*)(aPtr + k + 4);
        v2f b1 = *(const v2f*)(bPtr + k + 4);
        acc0 = __builtin_amdgcn_wmma_f32_16x16x4_f32(
            false, a0, false, b0, (short)0, acc0, false, false);
        acc1 = __builtin_amdgcn_wmma_f32_16x16x4_f32(
            false, a1, false, b1, (short)0, acc1, false, false);
    }

    // Diagonal extraction:
    //   lanes 0-7   own D[l][l]       in acc[l]
    //   lanes 24-31 own D[l-16][l-16] in acc[l-24]
    const int j = (lane < 16) ? lane : (lane - 24);   // valid for owning lanes
    float dot = 0.0f;
#pragma unroll
    for (int t = 0; t < 8; ++t) {
        const float c = acc0[t] + acc1[t];
        dot = (j == t) ? c : dot;
    }

    const bool own = (lane < 8) || (lane >= 24);
    const int myEdge = base + ((lane < 16) ? lane : (lane - 16));
    if (own && myEdge < E) {
        const float x = dot * w[0] + b[0];
        out[myEdge] = 1.0f / (1.0f + __expf(-x));
    }
}

extern "C" void kernel_launch(void* const* d_in, const int* in_sizes, int n_in,
                              void* d_out, int out_size, void* d_ws, size_t ws_size,
                              hipStream_t stream) {
    const float* Z     = (const float*)d_in[0];
    const int*   edges = (const int*)d_in[1];   // [2, E] int32 indices
    const float* w     = (const float*)d_in[2];
    const float* b     = (const float*)d_in[3];
    float*       out   = (float*)d_out;

    const int E = in_sizes[1] / 2;
    const int blocks = (E + EDGES_PER_BLOCK - 1) / EDGES_PER_BLOCK;

    edge_sigmoid_wmma<<<dim3(blocks), dim3(256), 0, stream>>>(Z, edges, w, b, out, E);
}